// SQNSurrogate_12275016532162
// MI455X (gfx1250) — compile-verified
//
#include <hip/hip_runtime.h>

// ---------------------------------------------------------------------------
// SNN pipeline for MI455X (gfx1250, wave32).  All GEMMs (3 convs as implicit
// GEMM, FC1 split-K, time-batched FC2/FC3) use V_WMMA_F32_16X16X4_F32 so
// arithmetic stays in f32 (spike thresholds are rounding-sensitive).  The
// T=10 LIF loop is factorized into per-layer elementwise time-scans +
// time-batched GEMMs, removing all cross-step synchronization.
// ---------------------------------------------------------------------------

typedef __attribute__((ext_vector_type(2))) float v2f;
typedef __attribute__((ext_vector_type(2), aligned(4))) float v2f_u; // 4B-aligned pair load
typedef __attribute__((ext_vector_type(8))) float v8f;

#define ALPHA 0.9f
#define BETA  0.8f
constexpr int TSIM   = 10;
constexpr int FC_IN  = 64 * 24 * 24 + 90;  // 36954
constexpr int KPAD   = 36960;              // FC_IN padded to multiple of 4
constexpr int KSPLIT = 4;                  // FC1 split-K factor
constexpr int KCHUNK = KPAD / KSPLIT;      // 9240 (multiple of 4)

static __device__ inline v8f wmma4(v2f a, v2f b, v8f c) {
  // D(16x16,f32) = A(16x4,f32) x B(4x16,f32) + C
  return __builtin_amdgcn_wmma_f32_16x16x4_f32(
      /*neg_a=*/false, a, /*neg_b=*/false, b,
      /*c_mod=*/(short)0, c, /*reuse_a=*/false, /*reuse_b=*/false);
}

// ---------------------------------------------------------------------------
// Implicit-GEMM conv + bias + ReLU.  One wave owns a 16(Cout) x (NB*16)(pos)
// tile of D[Cout, B*Hout*Wout]; the A (weight) fragment is reused across the
// NB N-tiles.  K = Cin*KH*KW stepped by 4 per WMMA.  For even KW the two
// im2col elements of a K-pair are adjacent in memory -> single v2f load.
// PACK=true writes into base[B, KPAD] feature layout (for conv3 -> FC1).
// Requires: Cout%16==0, K%4==0, (B*Hout*Wout)%(16*NB)==0.
// ---------------------------------------------------------------------------
template <int Cin, int KH, int KW, int S, int Hin, int Win,
          int Hout, int Wout, int Cout, bool PACK, int NB>
__global__ __launch_bounds__(256)
void conv_wmma(const float* __restrict__ in, const float* __restrict__ w,
               const float* __restrict__ bias, float* __restrict__ out, int B)
{
  constexpr int  K     = Cin * KH * KW;
  constexpr int  HW    = Hout * Wout;
  constexpr bool PAIRX = (KW % 2) == 0;   // K-pair always within one input row
  const int lane = threadIdx.x & 31;
  const int wave = threadIdx.x >> 5;
  const int nTilesM = Cout / 16;
  const long totalPos = (long)B * HW;
  const long nTiles = (totalPos / (16 * NB)) * nTilesM;
  const long tile = (long)blockIdx.x * 8 + wave;       // wave-uniform
  if (tile >= nTiles) return;                          // whole wave exits

  const int  tm     = (int)(tile % nTilesM);
  const long tnBase = (tile / nTilesM) * NB;
  const int  half = lane >> 4;
  const int  l15  = lane & 15;

  // per-lane output columns (positions) for the NB sub-tiles
  int bidx[NB], rem[NB], oyv[NB], oxv[NB];
  const float* __restrict__ inb[NB];
#pragma unroll
  for (int nb = 0; nb < NB; ++nb) {
    const long n = (tnBase + nb) * 16 + l15;
    bidx[nb] = (int)(n / HW);
    rem[nb]  = (int)(n % HW);
    oyv[nb]  = rem[nb] / Wout;
    oxv[nb]  = rem[nb] - oyv[nb] * Wout;
    inb[nb]  = in + (long)bidx[nb] * Cin * Hin * Win;
  }

  // this lane's A row (weight row); K-pair selected by half-wave
  const float* __restrict__ wrow = w + (long)(tm * 16 + l15) * K;

  v8f acc[NB];
#pragma unroll
  for (int nb = 0; nb < NB; ++nb) acc[nb] = (v8f){};

  for (int k0 = 0; k0 < K; k0 += 4) {
    const int ka = k0 + half * 2;
    v2f a;
    a.x = wrow[ka];
    a.y = wrow[ka + 1];
#pragma unroll
    for (int nb = 0; nb < NB; ++nb) {
      const int iy0 = oyv[nb] * S, ix0 = oxv[nb] * S;
      v2f bf;
      if (PAIRX) {
        // ka even, KW even -> kx and kx+1 in same input row, adjacent floats
        const int ci = ka / (KH * KW);
        const int r  = ka - ci * (KH * KW);
        const int ky = r / KW, kx = r - ky * KW;
        const v2f_u p = *(const v2f_u*)&inb[nb][((long)ci * Hin + (iy0 + ky)) * Win + (ix0 + kx)];
        bf.x = p.x; bf.y = p.y;
      } else {
        {
          const int ci = ka / (KH * KW);
          const int r  = ka - ci * (KH * KW);
          const int ky = r / KW, kx = r - ky * KW;
          bf.x = inb[nb][((long)ci * Hin + (iy0 + ky)) * Win + (ix0 + kx)];
        }
        {
          const int kb = ka + 1;
          const int ci = kb / (KH * KW);
          const int r  = kb - ci * (KH * KW);
          const int ky = r / KW, kx = r - ky * KW;
          bf.y = inb[nb][((long)ci * Hin + (iy0 + ky)) * Win + (ix0 + kx)];
        }
      }
      acc[nb] = wmma4(a, bf, acc[nb]);
    }
  }

  // C/D layout: VGPR v holds (M = v + 8*half, N = l15)
#pragma unroll
  for (int nb = 0; nb < NB; ++nb) {
#pragma unroll
    for (int v = 0; v < 8; ++v) {
      const int m = tm * 16 + v + 8 * half;
      float r = acc[nb][v] + bias[m];
      r = r > 0.f ? r : 0.f;
      if (PACK) {
        out[(long)bidx[nb] * KPAD + (long)m * HW + rem[nb]] = r;
      } else {
        out[(((long)bidx[nb] * Cout + m) * Hout + oyv[nb]) * Wout + oxv[nb]] = r;
      }
    }
  }
}

// ---------------------------------------------------------------------------
// base[:, FC_IN-90 : KPAD]: copy history (90) and zero the 6 pad columns.
// ---------------------------------------------------------------------------
__global__ void histpad_kernel(const float* __restrict__ hist,
                               float* __restrict__ base, int B)
{
  int i = blockIdx.x * blockDim.x + threadIdx.x;
  if (i >= B * 96) return;
  int b = i / 96, j = i - b * 96;
  base[(long)b * KPAD + (FC_IN - 90) + j] = (j < 90) ? hist[b * 90 + j] : 0.f;
}

// ---------------------------------------------------------------------------
// FC1 split-K partials: part[s][B,128] = base[:, sK:(s+1)K] @ w[:, sK:(s+1)K]^T.
// Deterministic: each split writes its own buffer; a fixed-order reduction
// kernel sums them (no FP atomics).  w index clamped in the tail chunk
// (padded base cols are 0, so clamped products contribute exactly 0).
// ---------------------------------------------------------------------------
__global__ __launch_bounds__(256)
void fc1_partial_kernel(const float* __restrict__ base, const float* __restrict__ w,
                        float* __restrict__ part, int B)
{
  const int lane = threadIdx.x & 31, wave = threadIdx.x >> 5;
  const int tilesPerSplit = (B / 16) * 8;                // N = 128 -> 8 tiles
  const int nTiles = tilesPerSplit * KSPLIT;
  const int tile = blockIdx.x * 8 + wave;
  if (tile >= nTiles) return;
  const int s  = tile / tilesPerSplit;
  const int r  = tile - s * tilesPerSplit;
  const int tm = r >> 3, tn = r & 7;
  const int half = lane >> 4, l15 = lane & 15;

  const float* __restrict__ arow = base + (long)(tm * 16 + l15) * KPAD;
  const int j = tn * 16 + l15;
  const float* __restrict__ brow = w + (long)j * FC_IN;

  const int kBeg = s * KCHUNK, kEnd = kBeg + KCHUNK;
  v8f acc = {};
  for (int k0 = kBeg; k0 < kEnd; k0 += 4) {
    const int ka = k0 + half * 2;
    v2f a;  a.x = arow[ka];  a.y = arow[ka + 1];
    const int k1 = (ka     < FC_IN) ? ka     : FC_IN - 1;
    const int k2 = (ka + 1 < FC_IN) ? ka + 1 : FC_IN - 1;
    v2f bf; bf.x = brow[k1]; bf.y = brow[k2];
    acc = wmma4(a, bf, acc);
  }
#pragma unroll
  for (int v = 0; v < 8; ++v) {
    const int row = tm * 16 + v + 8 * half;
    part[(long)s * (B * 128) + (long)row * 128 + j] = acc[v];
  }
}

__global__ void fc1_reduce_kernel(const float* __restrict__ part,
                                  const float* __restrict__ bias,
                                  float* __restrict__ h1, int n)
{
  int i = blockIdx.x * blockDim.x + threadIdx.x;
  if (i >= n) return;
  float acc = bias[i & 127];
#pragma unroll
  for (int s = 0; s < KSPLIT; ++s)        // fixed order -> deterministic
    acc += part[(long)s * n + i];
  h1[i] = acc;
}

// ---------------------------------------------------------------------------
// Time-batched FC2: h2[(t,b),256] = spk1[(t,b),128] @ fc2_w[256,128]^T + b.
// ---------------------------------------------------------------------------
__global__ __launch_bounds__(256)
void fc2_kernel(const float* __restrict__ spk, const float* __restrict__ w,
                const float* __restrict__ bias, float* __restrict__ h2, int Mrows)
{
  const int lane = threadIdx.x & 31, wave = threadIdx.x >> 5;
  const int nTiles = (Mrows / 16) * 16;                  // N = 256 -> 16 tiles
  const int tile = blockIdx.x * 8 + wave;
  if (tile >= nTiles) return;
  const int tm = tile >> 4, tn = tile & 15;
  const int half = lane >> 4, l15 = lane & 15;

  const float* __restrict__ arow = spk + (long)(tm * 16 + l15) * 128;
  const int j = tn * 16 + l15;
  const float* __restrict__ brow = w + (long)j * 128;

  v8f acc = {};
  for (int k0 = 0; k0 < 128; k0 += 4) {
    const int ka = k0 + half * 2;
    v2f a;  a.x = arow[ka];  a.y = arow[ka + 1];
    v2f bf; bf.x = brow[ka]; bf.y = brow[ka + 1];
    acc = wmma4(a, bf, acc);
  }
  const float bj = bias[j];
#pragma unroll
  for (int v = 0; v < 8; ++v) {
    const int row = tm * 16 + v + 8 * half;
    h2[(long)row * 256 + j] = acc[v] + bj;
  }
}

// ---------------------------------------------------------------------------
// Time-batched FC3: N = 9 padded to a 16-wide tile; B loads clamped, stores
// masked (divergence only after the last WMMA).
// ---------------------------------------------------------------------------
__global__ __launch_bounds__(256)
void fc3_kernel(const float* __restrict__ spk, const float* __restrict__ w,
                const float* __restrict__ bias, float* __restrict__ h3, int Mrows)
{
  const int lane = threadIdx.x & 31, wave = threadIdx.x >> 5;
  const int nTiles = Mrows / 16;
  const int tile = blockIdx.x * 8 + wave;
  if (tile >= nTiles) return;
  const int half = lane >> 4, l15 = lane & 15;

  const float* __restrict__ arow = spk + (long)(tile * 16 + l15) * 256;
  const int jc = (l15 < 9) ? l15 : 8;
  const float* __restrict__ brow = w + (long)jc * 256;

  v8f acc = {};
  for (int k0 = 0; k0 < 256; k0 += 4) {
    const int ka = k0 + half * 2;
    v2f a;  a.x = arow[ka];  a.y = arow[ka + 1];
    v2f bf; bf.x = brow[ka]; bf.y = brow[ka + 1];
    acc = wmma4(a, bf, acc);
  }
  if (l15 < 9) {
    const float bj = bias[l15];
#pragma unroll
    for (int v = 0; v < 8; ++v) {
      const int row = tile * 16 + v + 8 * half;
      h3[(long)row * 9 + l15] = acc[v] + bj;
    }
  }
}

// ---------------------------------------------------------------------------
// LIF scans (per-element private time loops; h1 is loop-invariant for L1).
// ---------------------------------------------------------------------------
__global__ void lif_scan_const(const float* __restrict__ h1,
                               float* __restrict__ spk_all, int n)
{
  int i = blockIdx.x * blockDim.x + threadIdx.x;
  if (i >= n) return;
  const float h = h1[i];
  float mem = 0.f, syn = 0.f;
#pragma unroll
  for (int t = 0; t < TSIM; ++t) {
    syn = ALPHA * syn + h;
    mem = BETA * mem + syn;
    const float s = (mem - 1.0f) > 0.0f ? 1.0f : 0.0f;
    spk_all[(long)t * n + i] = s;
    mem -= s;
  }
}

__global__ void lif_scan_seq(const float* __restrict__ h_all,
                             float* __restrict__ spk_all, int n)
{
  int i = blockIdx.x * blockDim.x + threadIdx.x;
  if (i >= n) return;
  float mem = 0.f, syn = 0.f;
#pragma unroll
  for (int t = 0; t < TSIM; ++t) {
    syn = ALPHA * syn + h_all[(long)t * n + i];
    mem = BETA * mem + syn;
    const float s = (mem - 1.0f) > 0.0f ? 1.0f : 0.0f;
    spk_all[(long)t * n + i] = s;
    mem -= s;
  }
}

__global__ void out_scan(const float* __restrict__ h3_all,
                         float* __restrict__ out, int n)
{
  int i = blockIdx.x * blockDim.x + threadIdx.x;
  if (i >= n) return;
  float mem = 0.f, syn = 0.f, pot = 0.f;
#pragma unroll
  for (int t = 0; t < TSIM; ++t) {
    syn = ALPHA * syn + h3_all[(long)t * n + i];
    mem = BETA * mem + syn;
    pot += mem;
  }
  out[i] = pot / (float)TSIM;
}

// ---------------------------------------------------------------------------
static inline int cdiv(long a, long b) { return (int)((a + b - 1) / b); }

extern "C" void kernel_launch(void* const* d_in, const int* in_sizes, int n_in,
                              void* d_out, int out_size, void* d_ws, size_t ws_size,
                              hipStream_t stream)
{
  const float* state   = (const float*)d_in[0];
  const float* history = (const float*)d_in[1];
  const float* c1w = (const float*)d_in[2];
  const float* c1b = (const float*)d_in[3];
  const float* c2w = (const float*)d_in[4];
  const float* c2b = (const float*)d_in[5];
  const float* c3w = (const float*)d_in[6];
  const float* c3b = (const float*)d_in[7];
  const float* f1w = (const float*)d_in[8];
  const float* f1b = (const float*)d_in[9];
  const float* f2w = (const float*)d_in[10];
  const float* f2b = (const float*)d_in[11];
  const float* f3w = (const float*)d_in[12];
  const float* f3b = (const float*)d_in[13];
  float* out = (float*)d_out;

  const int B = in_sizes[0] / (3 * 224 * 224);           // 256

  // ---- workspace layout (floats) ----
  float* ws = (float*)d_ws;
  const long n_c1  = (long)B * 32 * 55 * 55;             // conv1 out
  const long n_c2  = (long)B * 64 * 26 * 26;             // conv2 out
  const long n_bse = (long)B * KPAD;                     // packed conv3 + history
  float* conv1_out = ws;
  float* conv2_out = conv1_out + n_c1;
  float* base      = conv2_out + n_c2;
  float* h1        = base + n_bse;                       // [B,128]
  float* f1part    = h1 + (long)B * 128;                 // [KSPLIT][B,128]
  // conv1_out is dead after conv2 -> reuse for temporal buffers
  const long Mrows = (long)TSIM * B;                     // 2560
  float* spk1 = conv1_out;                               // [T*B,128]
  float* h2   = spk1 + Mrows * 128;                      // [T*B,256]
  float* spk2 = h2 + Mrows * 256;                        // [T*B,256]
  float* h3   = spk2 + Mrows * 256;                      // [T*B,9]

  // ---- conv stack (implicit GEMM, f32 WMMA, NB=2 register blocking) ----
  {
    const long tiles = ((long)B * 55 * 55 / 32) * (32 / 16);
    conv_wmma<3, 8, 8, 4, 224, 224, 55, 55, 32, false, 2>
        <<<cdiv(tiles, 8), 256, 0, stream>>>(state, c1w, c1b, conv1_out, B);
  }
  {
    const long tiles = ((long)B * 26 * 26 / 32) * (64 / 16);
    conv_wmma<32, 4, 4, 2, 55, 55, 26, 26, 64, false, 2>
        <<<cdiv(tiles, 8), 256, 0, stream>>>(conv1_out, c2w, c2b, conv2_out, B);
  }
  {
    const long tiles = ((long)B * 24 * 24 / 32) * (64 / 16);
    conv_wmma<64, 3, 3, 1, 26, 26, 24, 24, 64, true, 2>
        <<<cdiv(tiles, 8), 256, 0, stream>>>(conv2_out, c3w, c3b, base, B);
  }
  histpad_kernel<<<cdiv((long)B * 96, 256), 256, 0, stream>>>(history, base, B);

  // ---- FC1 (K = 36960 padded, deterministic split-K x4) ----
  {
    const long tiles = (long)(B / 16) * 8 * KSPLIT;
    fc1_partial_kernel<<<cdiv(tiles, 8), 256, 0, stream>>>(base, f1w, f1part, B);
    fc1_reduce_kernel<<<cdiv((long)B * 128, 256), 256, 0, stream>>>(f1part, f1b, h1, B * 128);
  }

  // ---- temporal loop, factorized ----
  lif_scan_const<<<cdiv((long)B * 128, 256), 256, 0, stream>>>(h1, spk1, B * 128);
  fc2_kernel<<<cdiv((Mrows / 16) * 16, 8), 256, 0, stream>>>(spk1, f2w, f2b, h2, (int)Mrows);
  lif_scan_seq<<<cdiv((long)B * 256, 256), 256, 0, stream>>>(h2, spk2, B * 256);
  fc3_kernel<<<cdiv(Mrows / 16, 8), 256, 0, stream>>>(spk2, f3w, f3b, h3, (int)Mrows);
  out_scan<<<cdiv((long)B * 9, 256), 256, 0, stream>>>(h3, out, B * 9);
}